// ExphormerAttention_11476152615031
// MI455X (gfx1250) — compile-verified
//
#include <hip/hip_runtime.h>
#include <hip/hip_bf16.h>

#define IN_DIM   128
#define OUT_DIM  128
#define NUM_HEADS 8
#define HEAD_DIM  16
#define LDS_PITCH 132   // 132 % 64 == 4 -> conflict-free half-wave A-fragment reads

typedef float v2f __attribute__((ext_vector_type(2)));
typedef float v8f __attribute__((ext_vector_type(8)));
typedef unsigned int v4u __attribute__((ext_vector_type(4)));
typedef int v8i __attribute__((ext_vector_type(8)));

// D = A(16x4 f32) * B(4x16 f32) + C(16x16 f32), full-precision WMMA.
static __device__ __forceinline__ v8f wmma_f32x4(v2f a, v2f b, v8f c) {
    return __builtin_amdgcn_wmma_f32_16x16x4_f32(
        /*neg_a=*/false, a, /*neg_b=*/false, b,
        /*c_mod=*/(short)0, c, /*reuse_a=*/false, /*reuse_b=*/false);
}

// ---------------------------------------------------------------------------
// Tensor Data Mover: DMA a [rows x 128] f32 tile (row stride 128 elems in
// memory) into LDS at lds_byte_off, padded to a 132-dword LDS pitch via the
// TDM pad feature (pad_interval=512B -> enc 6, pad_amount=16B -> enc 3).
// One issue per wave; completion tracked on TENSORcnt.
// ---------------------------------------------------------------------------
static __device__ __forceinline__ void tdm_load_tile_f32(const float* gsrc,
                                                         unsigned lds_byte_off,
                                                         int rows) {
    unsigned long long ga = (unsigned long long)(uintptr_t)gsrc;
    unsigned ga_lo = (unsigned)__builtin_amdgcn_readfirstlane((int)(unsigned)ga);
    unsigned ga_hi = (unsigned)__builtin_amdgcn_readfirstlane((int)(unsigned)(ga >> 32));
    unsigned lds   = (unsigned)__builtin_amdgcn_readfirstlane((int)lds_byte_off);
    unsigned nrows = (unsigned)__builtin_amdgcn_readfirstlane(rows);

    // D# group 0: count=1 | lds_addr | global_addr[56:0] | type=2
    v4u g0;
    g0.x = 1u;                                   // count=1, user mode, no gather
    g0.y = lds;                                  // lds_addr (bytes)
    g0.z = ga_lo;                                // global_addr[31:0]
    g0.w = (ga_hi & 0x01FFFFFFu) | (2u << 30);   // global_addr[56:32] | type=2

    // D# group 1:
    //  w0: wg_mask=0, data_size=2 (4B), pad_enable=1, pad_interval=6 (128 dw),
    //      pad_amount=3 (4 dw)  -> 0x07920000
    //  tensor_dim0=128, tensor_dim1=rows, tile_dim0=128, tile_dim1=16,
    //  tensor_dim0_stride=128
    v8i g1;
    g1[0] = (int)0x07920000u;
    g1[1] = (int)(128u << 16);                   // tensor_dim0[15:0] @ bits 63:48
    g1[2] = (int)(nrows << 16);                  // tensor_dim1[15:0] @ bits 95:80
    g1[3] = (int)(128u << 16);                   // tile_dim0 @ bits 127:112
    g1[4] = 16;                                  // tile_dim1 @ bits 143:128
    g1[5] = 128;                                 // tensor_dim0_stride[31:0]
    g1[6] = 0;
    g1[7] = 0;

    asm volatile("tensor_load_to_lds %0, %1" :: "s"(g0), "s"(g1) : "memory");
}

static __device__ __forceinline__ void tdm_wait() {
    asm volatile("s_wait_tensorcnt 0x0" ::: "memory");
}

__global__ void zero_kernel(float* __restrict__ p, size_t n) {
    size_t i = (size_t)blockIdx.x * blockDim.x + threadIdx.x;
    size_t stride = (size_t)gridDim.x * blockDim.x;
    for (; i < n; i += stride) p[i] = 0.0f;
}

// ---------------------------------------------------------------------------
// Node projections: Q = x@Wq+bq, K = x@Wk+bk, V = x@Wv+bv.
// Block = 128 threads (4 waves), handles 16 nodes. 24 (matrix, col-tile) jobs.
// x-tile staged by the TDM (wave 0 issues, block barrier publishes).
// ---------------------------------------------------------------------------
__global__ void proj_kernel(const float* __restrict__ x,
                            const float* __restrict__ Wq, const float* __restrict__ bq,
                            const float* __restrict__ Wk, const float* __restrict__ bk,
                            const float* __restrict__ Wv, const float* __restrict__ bv,
                            float* __restrict__ Q, float* __restrict__ K,
                            float* __restrict__ V, int n_nodes) {
    __shared__ float xa[16 * LDS_PITCH];

    const int wave = threadIdx.x >> 5;
    const int lane = threadIdx.x & 31;
    const int node0 = blockIdx.x * 16;
    const int rows = (n_nodes - node0 < 16) ? (n_nodes - node0) : 16;

    if (wave == 0) {
        tdm_load_tile_f32(x + (size_t)node0 * IN_DIM,
                          (unsigned)(uintptr_t)&xa[0], rows);
        tdm_wait();
    }
    __syncthreads();

    const int hh  = lane >> 4;   // half-wave: 0 or 1
    const int l16 = lane & 15;

    for (int j = wave; j < 24; j += 4) {
        const int mat = j >> 3;  // 0=Q 1=K 2=V
        const int t   = j & 7;   // 16-col tile
        const float* W = (mat == 0) ? Wq : (mat == 1) ? Wk : Wv;
        const float* b = (mat == 0) ? bq : (mat == 1) ? bk : bv;
        float* O       = (mat == 0) ? Q  : (mat == 1) ? K  : V;

        v8f acc = {};
        const int col = t * 16 + l16;
        for (int k = 0; k < IN_DIM; k += 4) {
            // A fragment: lanes 0-15 hold (M=l16, K=k..k+1), lanes 16-31 (K=k+2..k+3)
            const v2f a = *(const v2f*)(xa + l16 * LDS_PITCH + k + 2 * hh);
            v2f bb;
            bb.x = W[(size_t)(k + 2 * hh) * OUT_DIM + col];
            bb.y = W[(size_t)(k + 2 * hh + 1) * OUT_DIM + col];
            acc = wmma_f32x4(a, bb, acc);
        }
        const float bias = b[col];
#pragma unroll
        for (int r = 0; r < 8; ++r) {
            const int row = node0 + r + 8 * hh;   // C layout: lanes16-31 hold M=8..15
            if (row < n_nodes)
                O[(size_t)row * OUT_DIM + col] = acc[r] + bias;
        }
    }
}

// ---------------------------------------------------------------------------
// Edge kernel: per wave, 16 edges.
//  1) TDM-stage edge_attr tile to LDS (per-wave descriptor + TENSORcnt wait)
//  2) E = tile @ We + be via WMMA (E never touches HBM)
//  3) write E back to LDS in (edge, col) layout
//  4) per edge: coalesced K/Q/V row loads, per-head dot, exp(clamp), scatter
// ---------------------------------------------------------------------------
__global__ void edge_kernel(const float* __restrict__ edge_attr,
                            const int* __restrict__ edge_index,
                            const float* __restrict__ We, const float* __restrict__ be,
                            const float* __restrict__ Q, const float* __restrict__ K,
                            const float* __restrict__ V,
                            float* __restrict__ wV, float* __restrict__ Z,
                            int n_edges) {
    __shared__ float tiles[4][16 * LDS_PITCH];

    const int wave = threadIdx.x >> 5;
    const int lane = threadIdx.x & 31;
    const int e0 = (blockIdx.x * 4 + wave) * 16;
    if (e0 >= n_edges) return;
    float* xa = tiles[wave];

    const int nvalid = (n_edges - e0 < 16) ? (n_edges - e0) : 16;

    // Async DMA of the 16x128 edge_attr tile into this wave's LDS region.
    tdm_load_tile_f32(edge_attr + (size_t)e0 * IN_DIM,
                      (unsigned)(uintptr_t)xa, nvalid);

    const int hh  = lane >> 4;
    const int l16 = lane & 15;

    tdm_wait();

    // E = tile @ We : 8 column tiles, K=128 in steps of 4, A fragment reused.
    v8f acc[8] = {};
    for (int k = 0; k < IN_DIM; k += 4) {
        const v2f a = *(const v2f*)(xa + l16 * LDS_PITCH + k + 2 * hh);
        const float* w0 = We + (size_t)(k + 2 * hh) * OUT_DIM;
#pragma unroll
        for (int t = 0; t < 8; ++t) {
            v2f bb;
            bb.x = w0[t * 16 + l16];
            bb.y = w0[OUT_DIM + t * 16 + l16];
            acc[t] = wmma_f32x4(a, bb, acc[t]);
        }
    }

    // Bias + spill E back into the (now consumed) LDS tile, (edge,col) layout.
#pragma unroll
    for (int t = 0; t < 8; ++t) {
        const float bias = be[t * 16 + l16];
#pragma unroll
        for (int r = 0; r < 8; ++r) {
            xa[(r + 8 * hh) * LDS_PITCH + t * 16 + l16] = acc[t][r] + bias;
        }
    }

    // Score + scatter. lane covers cols 4*lane..4*lane+3, head = lane/4.
    const int head = lane >> 2;
    for (int e = 0; e < nvalid; ++e) {
        const int s = edge_index[e0 + e];             // src
        const int d = edge_index[n_edges + e0 + e];   // dst

        const float4 k4 = *(const float4*)(K + (size_t)s * OUT_DIM + lane * 4);
        const float4 q4 = *(const float4*)(Q + (size_t)d * OUT_DIM + lane * 4);
        const float4 e4 = *(const float4*)(xa + e * LDS_PITCH + lane * 4);

        float p = k4.x * q4.x * e4.x + k4.y * q4.y * e4.y +
                  k4.z * q4.z * e4.z + k4.w * q4.w * e4.w;
        // reduce across the 4 lanes of this head
        p += __shfl_xor(p, 1);
        p += __shfl_xor(p, 2);

        float sc = p * 0.25f;                         // 1/sqrt(HEAD_DIM)
        sc = fminf(5.0f, fmaxf(-5.0f, sc));
        const float score = __expf(sc);

        const float4 v4 = *(const float4*)(V + (size_t)s * OUT_DIM + lane * 4);
        float* dp = wV + (size_t)d * OUT_DIM + lane * 4;
        atomicAdd(dp + 0, v4.x * score);
        atomicAdd(dp + 1, v4.y * score);
        atomicAdd(dp + 2, v4.z * score);
        atomicAdd(dp + 3, v4.w * score);
        if ((lane & 3) == 0) atomicAdd(Z + (size_t)d * NUM_HEADS + head, score);
    }
}

__global__ void final_kernel(float* __restrict__ out, const float* __restrict__ Z,
                             int n_nodes) {
    const size_t i = (size_t)blockIdx.x * blockDim.x + threadIdx.x;
    const size_t total = (size_t)n_nodes * OUT_DIM;
    if (i < total) {
        const int node = (int)(i >> 7);
        const int col  = (int)(i & 127);
        out[i] = out[i] / (Z[(size_t)node * NUM_HEADS + (col >> 4)] + 1e-6f);
    }
}

extern "C" void kernel_launch(void* const* d_in, const int* in_sizes, int n_in,
                              void* d_out, int out_size, void* d_ws, size_t ws_size,
                              hipStream_t stream) {
    const float* x         = (const float*)d_in[0];
    const float* edge_attr = (const float*)d_in[1];
    const int*   edge_idx  = (const int*)d_in[2];
    const float* Wq = (const float*)d_in[3];
    const float* bq = (const float*)d_in[4];
    const float* Wk = (const float*)d_in[5];
    const float* bk = (const float*)d_in[6];
    const float* We = (const float*)d_in[7];
    const float* be = (const float*)d_in[8];
    const float* Wv = (const float*)d_in[9];
    const float* bv = (const float*)d_in[10];
    float* out = (float*)d_out;

    const int n_nodes = in_sizes[0] / IN_DIM;
    const int n_edges = in_sizes[1] / IN_DIM;

    float* ws = (float*)d_ws;
    float* Q = ws;
    float* K = Q + (size_t)n_nodes * OUT_DIM;
    float* V = K + (size_t)n_nodes * OUT_DIM;
    float* Z = V + (size_t)n_nodes * OUT_DIM;
    // total ws use: 3*N*128 + N*8 floats  (~78.4 MB for N=50000)

    // wV is accumulated directly in d_out; zero it and Z.
    zero_kernel<<<2048, 256, 0, stream>>>(out, (size_t)n_nodes * OUT_DIM);
    zero_kernel<<<256, 256, 0, stream>>>(Z, (size_t)n_nodes * NUM_HEADS);

    proj_kernel<<<(n_nodes + 15) / 16, 128, 0, stream>>>(
        x, Wq, bq, Wk, bk, Wv, bv, Q, K, V, n_nodes);

    edge_kernel<<<(n_edges + 63) / 64, 128, 0, stream>>>(
        edge_attr, edge_idx, We, be, Q, K, V, out, Z, n_edges);

    final_kernel<<<((int)(((size_t)n_nodes * OUT_DIM + 255) / 256)), 256, 0, stream>>>(
        out, Z, n_nodes);
}